// General_detector_89876485636347
// MI455X (gfx1250) — compile-verified
//
#include <hip/hip_runtime.h>
#include <hip/hip_bf16.h>
#include <math.h>
#include <stdint.h>

#define N_ANCH   785664
#define NCLS     80
#define PRE_K    1000
#define PAD_K    1008     /* 63 * 16 */
#define NTILE    63
#define MAX_DET  100
#define HIST_B   8192
#define CAP      4096
#define SCORE_THRESH 0.05f
#define IOU_THRESH   0.5f

typedef __attribute__((ext_vector_type(2))) float v2f;
typedef __attribute__((ext_vector_type(8))) float v8f;

// ---------------------------------------------------------------- init
__global__ __launch_bounds__(256)
void init_kernel(int* hist, int* counters, unsigned* masks) {
    int t = blockIdx.x * blockDim.x + threadIdx.x;
    if (t < HIST_B) hist[t] = 0;
    if (t < 16)     counters[t] = 0;
    if (t < PAD_K * 32) masks[t] = 0u;
}

// --------------------------------------------- decode + clip + cls max
__global__ __launch_bounds__(256)
void decode_score_kernel(const float* __restrict__ reg,
                         const float* __restrict__ cls,
                         const float* __restrict__ anc,
                         const int* __restrict__ imgw,
                         const int* __restrict__ imgh,
                         float* __restrict__ boxes,
                         float* __restrict__ scores,
                         int*   __restrict__ labels)
{
    int tid = blockIdx.x * blockDim.x + threadIdx.x;
    if (tid >= N_ANCH) return;
    float4 r4 = ((const float4*)reg)[tid];
    float4 a4 = ((const float4*)anc)[tid];
    float W = (float)imgw[0], H = (float)imgh[0];
    float wa = a4.z - a4.x, ha = a4.w - a4.y;
    float cxa = a4.x + 0.5f * wa, cya = a4.y + 0.5f * ha;
    float dx = r4.x * 0.1f, dy = r4.y * 0.1f;
    float dw = r4.z * 0.2f, dh = r4.w * 0.2f;
    float cx = cxa + dx * wa, cy = cya + dy * ha;
    float w = __expf(dw) * wa, h = __expf(dh) * ha;
    float x1 = fminf(fmaxf(cx - 0.5f * w, 0.f), W);
    float y1 = fminf(fmaxf(cy - 0.5f * h, 0.f), H);
    float x2 = fminf(fmaxf(cx + 0.5f * w, 0.f), W);
    float y2 = fminf(fmaxf(cy + 0.5f * h, 0.f), H);
    ((float4*)boxes)[tid] = make_float4(x1, y1, x2, y2);

    const float4* cp = (const float4*)(cls + (size_t)tid * NCLS);
    __builtin_prefetch(cp + 20 * 64, 0, 0);   // stream-ahead hint -> global_prefetch_b8
    float m = -1e30f; int mi = 0;
#pragma unroll
    for (int k = 0; k < NCLS / 4; ++k) {
        float4 v = cp[k];
        if (v.x > m) { m = v.x; mi = 4 * k + 0; }
        if (v.y > m) { m = v.y; mi = 4 * k + 1; }
        if (v.z > m) { m = v.z; mi = 4 * k + 2; }
        if (v.w > m) { m = v.w; mi = 4 * k + 3; }
    }
    float sc = 1.0f / (1.0f + __expf(-m));   // max(sigmoid) == sigmoid(max)
    if (!(sc > SCORE_THRESH)) sc = -1.0f;
    scores[tid] = sc;
    labels[tid] = mi;
}

// ---------------------------------------------------------- histogram
__global__ __launch_bounds__(256)
void hist_kernel(const float* __restrict__ scores, int* __restrict__ hist) {
    int t = blockIdx.x * blockDim.x + threadIdx.x;
    if (t >= N_ANCH) return;
    float s = scores[t];
    if (s > 0.f) {
        int b = (int)(s * (float)HIST_B);
        b = min(max(b, 0), HIST_B - 1);
        atomicAdd(&hist[b], 1);
    }
}

// ------------------------------------------------- threshold-bucket scan
__global__ void thresh_kernel(const int* __restrict__ hist, int* counters) {
    if (threadIdx.x == 0 && blockIdx.x == 0) {
        int cum = 0, T = 0;
        for (int b = HIST_B - 1; b >= 0; --b) {
            cum += hist[b];
            if (cum >= PRE_K) { T = b; break; }
        }
        counters[0] = T;
    }
}

// ---------------------------------------------------------- compaction
__global__ __launch_bounds__(256)
void compact_kernel(const float* __restrict__ scores,
                    int* counters, float* cand_s, int* cand_i)
{
    int t = blockIdx.x * blockDim.x + threadIdx.x;
    if (t >= N_ANCH) return;
    float s = scores[t];
    if (s > 0.f) {
        int b = (int)(s * (float)HIST_B);
        b = min(max(b, 0), HIST_B - 1);
        if (b >= counters[0]) {
            int pos = atomicAdd(&counters[2], 1);
            if (pos < CAP) { cand_s[pos] = s; cand_i[pos] = t; }
        }
    }
}

// ----------------------------------- bitonic sort of candidates (desc)
__global__ __launch_bounds__(1024)
void sort_cand_kernel(const float* __restrict__ cand_s, const int* __restrict__ cand_i,
                      const int* __restrict__ counters,
                      const float* __restrict__ boxes, const int* __restrict__ labels,
                      float* __restrict__ top_scores, int* __restrict__ top_labels,
                      float* __restrict__ top_boxes, float* __restrict__ off_boxes)
{
    __shared__ float ss[CAP];
    __shared__ int   si[CAP];
    int tid = threadIdx.x;
    int cnt = counters[2]; if (cnt > CAP) cnt = CAP;
    for (int i = tid; i < CAP; i += 1024) {
        if (i < cnt) { ss[i] = cand_s[i]; si[i] = cand_i[i]; }
        else         { ss[i] = -1.0f;     si[i] = 0x7FFFFFFF; }
    }
    __syncthreads();
    for (int k = 2; k <= CAP; k <<= 1) {
        for (int j = k >> 1; j > 0; j >>= 1) {
            for (int i = tid; i < CAP; i += 1024) {
                int ixj = i ^ j;
                if (ixj > i) {
                    bool descBlk = ((i & k) == 0);
                    float a = ss[i], b = ss[ixj];
                    int  ia = si[i], ib = si[ixj];
                    bool aFirst = (a > b) || (a == b && ia < ib);
                    if (descBlk ? !aFirst : aFirst) {
                        ss[i] = b; ss[ixj] = a; si[i] = ib; si[ixj] = ia;
                    }
                }
            }
            __syncthreads();
        }
    }
    for (int t = tid; t < PAD_K; t += 1024) {
        float sc = -1.f; int gi = -1;
        if (t < PRE_K && t < cnt) { sc = ss[t]; gi = si[t]; }
        bool valid = (gi >= 0 && gi < N_ANCH);
        if (valid) {
            float4 b = ((const float4*)boxes)[gi];
            int lb = labels[gi];
            top_scores[t] = sc;
            top_labels[t] = lb;
            ((float4*)top_boxes)[t] = b;
            float off = (float)lb * 2049.0f;   // max(img_w,img_h)+1
            ((float4*)off_boxes)[t] = make_float4(b.x + off, b.y + off, b.z + off, b.w + off);
        } else {
            top_scores[t] = -1.f;
            top_labels[t] = 0;
            ((float4*)top_boxes)[t] = make_float4(0.f, 0.f, 0.f, 0.f);
            ((float4*)off_boxes)[t] = make_float4(-1e8f, -1e8f, -1e8f, -1e8f);
        }
    }
    if (tid >= PAD_K && tid < 1024) top_scores[tid] = -1.f;
}

// ------------------- pairwise IoU tile kernel, WMMA for area_i + area_j
// One wave per 16x16 tile. Boxes staged into LDS with async-to-LDS DMA.
__global__ __launch_bounds__(32)
void iou_mask_kernel(const float* __restrict__ off_boxes, unsigned* __restrict__ masks)
{
    __shared__ float4 sb[32];                            // [0:16)=row boxes, [16:32)=col boxes
    int bj = blockIdx.x, bi = blockIdx.y;
    int l = threadIdx.x;
    bool lo = l < 16;
    int l15 = l & 15;
    const float4* ob = (const float4*)off_boxes;

    // lanes 0-15 fetch this tile's row boxes, lanes 16-31 its column boxes
    int srcIdx = lo ? (bi * 16 + l15) : (bj * 16 + l15);
    const float4* gsrc = ob + srcIdx;
    unsigned ldsoff = (unsigned)(uintptr_t)(&sb[l]);     // addr[31:0] == LDS offset (ISA aperture map)
    asm volatile("global_load_async_to_lds_b128 %0, %1, off\n\t"
                 "s_wait_asynccnt 0"
                 :: "v"(ldsoff), "v"(gsrc) : "memory");
    __syncthreads();                                     // single-wave WG: lowers to S_NOP

    float4 cb  = sb[16 + l15];                           // this lane's column box
    float4 rbA = sb[l15];                                // row box for A operand
    float areaC = (cb.z - cb.x) * (cb.w - cb.y);
    float areaR = (rbA.z - rbA.x) * (rbA.w - rbA.y);

    // A (16x4): A[M,0]=areaR[M], A[M,1]=1 ; B (4x16): B[0,N]=1, B[1,N]=areaC[N]
    v2f A, B;
    A[0] = lo ? areaR : 0.f;  A[1] = lo ? 1.f : 0.f;
    B[0] = lo ? 1.f : 0.f;    B[1] = lo ? areaC : 0.f;

    float inter[8];
    v8f C;
    int hi8 = lo ? 0 : 8;
#pragma unroll
    for (int r = 0; r < 8; ++r) {
        float4 rb = sb[r + hi8];                         // LDS broadcast within half-wave
        float iw = fminf(rb.z, cb.z) - fmaxf(rb.x, cb.x);
        float ih = fminf(rb.w, cb.w) - fmaxf(rb.y, cb.y);
        iw = fmaxf(iw, 0.f); ih = fmaxf(ih, 0.f);
        inter[r] = iw * ih;
        C[r] = -inter[r];
    }
    // D[M][N] = area_M + area_N - inter  (tensor op, C/D lane layout per ISA)
    v8f D = __builtin_amdgcn_wmma_f32_16x16x4_f32(false, A, false, B, (short)0, C, false, false);

    unsigned bal[8];
#pragma unroll
    for (int r = 0; r < 8; ++r) {
        // iou > 0.5  <=>  inter > 0.5*denom   (denom > 0 always) -- no division
        bool hit = inter[r] > 0.5f * (D[r] + 1e-8f);
        bal[r] = (unsigned)__ballot(hit);
    }
    if (l == 0) {
        int word = bj >> 1;
        unsigned sh = (unsigned)((bj & 1) * 16);
#pragma unroll
        for (int r = 0; r < 8; ++r) {
            atomicOr(&masks[(size_t)(bi * 16 + r)     * 32 + word], (bal[r] & 0xFFFFu) << sh);
            atomicOr(&masks[(size_t)(bi * 16 + r + 8) * 32 + word], ((bal[r] >> 16) & 0xFFFFu) << sh);
        }
    }
}

// ----------------------------------------------- sequential greedy NMS
__global__ __launch_bounds__(32)
void nms_scan_kernel(const float* __restrict__ top_scores,
                     const unsigned* __restrict__ masks,
                     float* __restrict__ final_scores)
{
    __shared__ unsigned removed[32];
    int lane = threadIdx.x;
    removed[lane] = 0u;
    __syncthreads();
    for (int i = 0; i < PRE_K; ++i) {
        unsigned rw = removed[i >> 5];                   // uniform across wave
        bool alive = (((rw >> (i & 31)) & 1u) == 0u) && (top_scores[i] > 0.0f);
        __syncthreads();
        if (alive) removed[lane] |= masks[(size_t)i * 32 + lane];
        if (lane == 0) final_scores[i] = alive ? top_scores[i] : -1.0f;
        __syncthreads();
    }
    for (int i = PRE_K + lane; i < 1024; i += 32) final_scores[i] = -1.0f;
}

// ------------------------------------------------- final top-100 output
__global__ __launch_bounds__(1024)
void final_topk_kernel(const float* __restrict__ final_scores,
                       const int* __restrict__ top_labels,
                       const float* __restrict__ top_boxes,
                       float* __restrict__ out)
{
    __shared__ float fs[1024];
    __shared__ int   fi[1024];
    int tid = threadIdx.x;
    fs[tid] = final_scores[tid];
    fi[tid] = tid;
    __syncthreads();
    for (int k = 2; k <= 1024; k <<= 1) {
        for (int j = k >> 1; j > 0; j >>= 1) {
            int ixj = tid ^ j;
            if (ixj > tid) {
                bool descBlk = ((tid & k) == 0);
                float a = fs[tid], b = fs[ixj];
                int  ia = fi[tid], ib = fi[ixj];
                bool aFirst = (a > b) || (a == b && ia < ib);
                if (descBlk ? !aFirst : aFirst) {
                    fs[tid] = b; fs[ixj] = a; fi[tid] = ib; fi[ixj] = ia;
                }
            }
            __syncthreads();
        }
    }
    if (tid < MAX_DET) {
        float sc = fs[tid];
        int s = fi[tid];
        float lb = 0.f;
        float4 bx = make_float4(0.f, 0.f, 0.f, 0.f);
        if (s < PAD_K) { lb = (float)top_labels[s]; bx = ((const float4*)top_boxes)[s]; }
        out[tid] = sc;                      // scores [0,100)
        out[MAX_DET + tid] = lb;            // labels [100,200)
        out[2 * MAX_DET + tid * 4 + 0] = bx.x;   // boxes [200,600)
        out[2 * MAX_DET + tid * 4 + 1] = bx.y;
        out[2 * MAX_DET + tid * 4 + 2] = bx.z;
        out[2 * MAX_DET + tid * 4 + 3] = bx.w;
    }
}

// ---------------------------------------------------------------- host
extern "C" void kernel_launch(void* const* d_in, const int* in_sizes, int n_in,
                              void* d_out, int out_size, void* d_ws, size_t ws_size,
                              hipStream_t stream) {
    const float* reg  = (const float*)d_in[0];
    const float* cls  = (const float*)d_in[1];
    const float* anc  = (const float*)d_in[2];
    const int*   imgw = (const int*)d_in[3];
    const int*   imgh = (const int*)d_in[4];
    float* out = (float*)d_out;

    char* ws = (char*)d_ws;
    size_t o = 0;
    auto take = [&](size_t bytes) -> void* {
        void* p = ws + o;
        o += (bytes + 255) & ~(size_t)255;
        return p;
    };
    float*    boxes        = (float*)take((size_t)N_ANCH * 4 * sizeof(float));
    float*    scores       = (float*)take((size_t)N_ANCH * sizeof(float));
    int*      labels       = (int*)take((size_t)N_ANCH * sizeof(int));
    int*      hist         = (int*)take(HIST_B * sizeof(int));
    int*      counters     = (int*)take(16 * sizeof(int));
    float*    cand_s       = (float*)take(CAP * sizeof(float));
    int*      cand_i       = (int*)take(CAP * sizeof(int));
    float*    top_scores   = (float*)take(1024 * sizeof(float));
    int*      top_labels   = (int*)take(1024 * sizeof(int));
    float*    top_boxes    = (float*)take(1024 * 4 * sizeof(float));
    float*    off_boxes    = (float*)take(1024 * 4 * sizeof(float));
    unsigned* masks        = (unsigned*)take((size_t)PAD_K * 32 * sizeof(unsigned));
    float*    final_scores = (float*)take(1024 * sizeof(float));

    int nblk = (N_ANCH + 255) / 256;
    int zblk = (PAD_K * 32 + 255) / 256;

    init_kernel<<<zblk, 256, 0, stream>>>(hist, counters, masks);
    decode_score_kernel<<<nblk, 256, 0, stream>>>(reg, cls, anc, imgw, imgh,
                                                  boxes, scores, labels);
    hist_kernel<<<nblk, 256, 0, stream>>>(scores, hist);
    thresh_kernel<<<1, 32, 0, stream>>>(hist, counters);
    compact_kernel<<<nblk, 256, 0, stream>>>(scores, counters, cand_s, cand_i);
    sort_cand_kernel<<<1, 1024, 0, stream>>>(cand_s, cand_i, counters, boxes, labels,
                                             top_scores, top_labels, top_boxes, off_boxes);
    iou_mask_kernel<<<dim3(NTILE, NTILE), 32, 0, stream>>>(off_boxes, masks);
    nms_scan_kernel<<<1, 32, 0, stream>>>(top_scores, masks, final_scores);
    final_topk_kernel<<<1, 1024, 0, stream>>>(final_scores, top_labels, top_boxes, out);
}